// Linear_71090298683714
// MI455X (gfx1250) — compile-verified
//
#include <hip/hip_runtime.h>
#include <math.h>

#define N_IN      1034      // 1024 x-features + 10 pulses
#define N_X       1024
#define NPULSE    10
#define OUTF      512
#define BATCH     32
#define PADN      2048      // bitonic sort padding
#define CHUNK     32
#define NCHUNK    33        // ceil(1034/32)
#define NBOUND    32        // carry boundaries c=1..32 (rank < 32*c)
#define KP        1040      // padded K (multiple of 16): clean, branch-free GEMM loop
#define MROWS     (BATCH * NBOUND)   // 1024
#define KNS       1000.0f   // K_NO_SPIKE
#define MIN_W_ARG -0.3678794411714423f

typedef __attribute__((ext_vector_type(2))) float v2f;
typedef __attribute__((ext_vector_type(8))) float v8f;

// ---------------- Lambert W0 on [-1/e, 0], Halley iterations ----------------
__device__ __forceinline__ float lambertw0f(float x) {
    const float e = 2.718281828459045f;
    float w = (x < -0.25f)
                  ? (-1.0f + sqrtf(fmaxf(2.0f * (1.0f + e * x), 0.0f)))
                  : x * (1.0f - x);
#pragma unroll
    for (int i = 0; i < 8; ++i) {
        float ew = __expf(w);
        float f  = w * ew - x;
        float denom = ew * (w + 1.0f) - (w + 2.0f) * f / (2.0f * (w + 1.0f) + 1e-12f);
        w = w - f / (denom + 1e-12f);
    }
    return w;
}

// ---------------- Kernel 1: per-batch bitonic sort + aux arrays -------------
__global__ void sort_prep_kernel(const float* __restrict__ x,
                                 const float* __restrict__ pulse,
                                 float* __restrict__ t_s, int* __restrict__ ord,
                                 int* __restrict__ rank,
                                 float* __restrict__ zv, float* __restrict__ ztv) {
    __shared__ float skey[PADN];
    __shared__ int   sidx[PADN];
    const int b = blockIdx.x;
    const int tid = threadIdx.x;

    for (int i = tid; i < PADN; i += blockDim.x) {
        float v;
        if (i < N_X)        v = x[b * N_X + i];
        else if (i < N_IN)  v = pulse[i - N_X];
        else                v = 3.0e38f;          // pad sorts past all real times
        skey[i] = v;
        sidx[i] = i;
    }
    __syncthreads();

    for (unsigned k = 2; k <= PADN; k <<= 1) {
        for (unsigned j = k >> 1; j > 0; j >>= 1) {
            for (unsigned i = tid; i < PADN; i += blockDim.x) {
                unsigned ixj = i ^ j;
                if (ixj > i) {
                    bool up = ((i & k) == 0);
                    float a = skey[i], c2 = skey[ixj];
                    bool doswap = up ? (a > c2) : (a < c2);
                    if (doswap) {
                        skey[i] = c2; skey[ixj] = a;
                        int tmp = sidx[i]; sidx[i] = sidx[ixj]; sidx[ixj] = tmp;
                    }
                }
            }
            __syncthreads();
        }
    }

    for (int s = tid; s < N_IN; s += blockDim.x) {
        float t = skey[s];
        int   n = sidx[s];
        t_s[b * N_IN + s] = t;
        ord[b * N_IN + s] = n;
        rank[b * N_IN + n] = s;                   // position of input n in sorted order
    }
    for (int n = tid; n < N_IN; n += blockDim.x) {
        float a = (n < N_X) ? x[b * N_X + n] : pulse[n - N_X];
        float z = (a < KNS) ? __expf(a) : 0.0f;   // DECAY_RATE = 1
        zv[b * N_IN + n]  = z;
        ztv[b * N_IN + n] = z * a;
    }
}

// ---------------- Kernel 2a: build masked, padded GEMM operands -------------
// ZA[(b*32 + c-1), n] = z[b,n]  * [rank(b,n) < 32*c]    (zero tail n>=1034)
// ZB[(b*32 + c-1), n] = zt[b,n] * [rank(b,n) < 32*c]
__global__ void build_z_kernel(const int* __restrict__ rank,
                               const float* __restrict__ zv,
                               const float* __restrict__ ztv,
                               float* __restrict__ ZA, float* __restrict__ ZB) {
    int idx = blockIdx.x * blockDim.x + threadIdx.x;
    if (idx >= MROWS * KP) return;
    const int n = idx % KP;
    const int r = idx / KP;
    const int b = r >> 5;
    const int boundary = ((r & 31) + 1) * CHUNK;
    float za = 0.0f, zb = 0.0f;
    if (n < N_IN && rank[b * N_IN + n] < boundary) {
        za = zv[b * N_IN + n];
        zb = ztv[b * N_IN + n];
    }
    ZA[idx] = za;
    ZB[idx] = zb;
}

// ---------------- Kernel 2b: zero-padded weight copy ------------------------
__global__ void build_w_kernel(const float* __restrict__ W, float* __restrict__ Wp) {
    int idx = blockIdx.x * blockDim.x + threadIdx.x;
    if (idx >= OUTF * KP) return;
    const int n = idx % KP;
    const int o = idx / KP;
    Wp[idx] = (n < N_IN) ? W[o * N_IN + n] : 0.0f;
}

// ---------------- Kernel 3: carry GEMM via V_WMMA_F32_16X16X4_F32 -----------
// Branch-free inner loop: 3x global_load_b64 + 2x v_wmma per K-step.
__global__ void prefix_gemm_kernel(const float* __restrict__ Wp,
                                   const float* __restrict__ ZA,
                                   const float* __restrict__ ZB,
                                   float* __restrict__ prefA,
                                   float* __restrict__ prefB) {
    const int lane = threadIdx.x & 31;
    const int wave = threadIdx.x >> 5;
    const int tile = blockIdx.x * (blockDim.x >> 5) + wave;
    const int NT = OUTF / 16;               // 32 tiles along o
    const int mt = tile / NT, nt = tile % NT;
    const int row0 = mt * 16, col0 = nt * 16;

    // A-fragment (16x4 f32): lanes 0-15 -> M=lane, K={0,1}; lanes 16-31 -> K={2,3}
    const int mA   = lane & 15;
    const int kg   = lane >> 4;             // K sub-group: elements 2*kg, 2*kg+1
    const int r    = row0 + mA;             // global M row = b*32 + (c-1)
    const int nCol = col0 + (lane & 15);    // output-neuron column (B fragment)

    const float* pA = ZA + (size_t)r * KP + 2 * kg;
    const float* pB = ZB + (size_t)r * KP + 2 * kg;
    const float* pW = Wp + (size_t)nCol * KP + 2 * kg;

    v8f accA = {0.f, 0.f, 0.f, 0.f, 0.f, 0.f, 0.f, 0.f};
    v8f accB = {0.f, 0.f, 0.f, 0.f, 0.f, 0.f, 0.f, 0.f};

#pragma unroll 4
    for (int k0 = 0; k0 < KP; k0 += 4) {
        v2f aA = *(const v2f*)(pA + k0);    // aligned 8B -> global_load_b64
        v2f aB = *(const v2f*)(pB + k0);
        v2f bw = *(const v2f*)(pW + k0);
        accA = __builtin_amdgcn_wmma_f32_16x16x4_f32(false, aA, false, bw,
                                                     (short)0, accA, false, false);
        accB = __builtin_amdgcn_wmma_f32_16x16x4_f32(false, aB, false, bw,
                                                     (short)0, accB, false, false);
    }

    // C/D layout: VGPR rr -> M = rr + (lane>=16 ? 8 : 0), N = lane&15
    const int mHi = (lane >> 4) * 8;
#pragma unroll
    for (int rr = 0; rr < 8; ++rr) {
        const int gRow = row0 + mHi + rr;
        const int gCol = col0 + (lane & 15);
        prefA[gRow * OUTF + gCol] = accA[rr];
        prefB[gRow * OUTF + gCol] = accB[rr];
    }
}

// ---------------- Kernel 4: init packed first-fire slots --------------------
__global__ void init_kernel(unsigned long long* __restrict__ packed) {
    int i = blockIdx.x * blockDim.x + threadIdx.x;
    if (i >= BATCH * OUTF) return;
    union { float f; unsigned u; } cv; cv.f = KNS;
    packed[i] = (0xFFFFFFFFull << 32) | cv.u;     // sentinel: no fire
}

// ---------------- Kernel 5: parallel per-chunk refinement -------------------
__global__ void refine_kernel(const float* __restrict__ W,
                              const float* __restrict__ t_s,
                              const int* __restrict__ ord,
                              const float* __restrict__ prefA,
                              const float* __restrict__ prefB,
                              unsigned long long* __restrict__ packed) {
    int idx = blockIdx.x * blockDim.x + threadIdx.x;
    if (idx >= BATCH * OUTF * NCHUNK) return;
    const int c = idx % NCHUNK;
    const int o = (idx / NCHUNK) % OUTF;
    const int b = idx / (NCHUNK * OUTF);

    float A = 0.0f, Bv = 0.0f;
    if (c > 0) {
        const int prow = b * NBOUND + (c - 1);
        A  = prefA[prow * OUTF + o];
        Bv = prefB[prow * OUTF + o];
    }
    const int s0 = c * CHUNK;
    const int s1 = (s0 + CHUNK < N_IN) ? s0 + CHUNK : N_IN;

    for (int s = s0; s < s1; ++s) {
        const float t = t_s[b * N_IN + s];
        if (t >= KNS) break;                       // valid_in fails from here on
        const float z = __expf(t);
        const float w = W[o * N_IN + ord[b * N_IN + s]];
        A  += w * z;
        Bv += w * z * t;
        if (A > 1e-10f) {
            const float boa = Bv / A;
            const float arg = -(1.0f / A) * __expf(fminf(boa, 80.0f));
            if (arg >= MIN_W_ARG) {
                const float wl = lambertw0f(fminf(fmaxf(arg, MIN_W_ARG), 0.0f));
                const float tc = boa - wl;
                const float tn = (s + 1 < N_IN) ? t_s[b * N_IN + s + 1] : KNS;
                if (tc >= t && tc <= tn) {
                    union { float f; unsigned u; } cv; cv.f = tc;
                    unsigned long long key =
                        ((unsigned long long)(unsigned)s << 32) | cv.u;
                    atomicMin(&packed[b * OUTF + o], key);
                    break;                          // first valid in this chunk
                }
            }
        }
    }
}

// ---------------- Kernel 6: decode packed -> spike_time ---------------------
__global__ void finalize_kernel(const unsigned long long* __restrict__ packed,
                                float* __restrict__ out) {
    int i = blockIdx.x * blockDim.x + threadIdx.x;
    if (i >= BATCH * OUTF) return;
    const unsigned long long v = packed[i];
    const unsigned hi = (unsigned)(v >> 32);
    union { unsigned u; float f; } cv; cv.u = (unsigned)v;
    out[i] = (hi == 0xFFFFFFFFu) ? KNS : cv.f;
}

extern "C" void kernel_launch(void* const* d_in, const int* in_sizes, int n_in,
                              void* d_out, int out_size, void* d_ws, size_t ws_size,
                              hipStream_t stream) {
    const float* x     = (const float*)d_in[0];   // [32, 1024]
    const float* W     = (const float*)d_in[1];   // [512, 1034]
    const float* pulse = (const float*)d_in[2];   // [10]
    float* out = (float*)d_out;                   // [32, 512]
    (void)in_sizes; (void)n_in; (void)out_size; (void)ws_size;

    char* ws = (char*)d_ws;
    size_t off = 0;
    unsigned long long* packed = (unsigned long long*)(ws + off); off += (size_t)BATCH * OUTF * 8;
    float* t_s   = (float*)(ws + off); off += (size_t)BATCH * N_IN * 4;
    int*   ord   = (int*)  (ws + off); off += (size_t)BATCH * N_IN * 4;
    int*   rank  = (int*)  (ws + off); off += (size_t)BATCH * N_IN * 4;
    float* zv    = (float*)(ws + off); off += (size_t)BATCH * N_IN * 4;
    float* ztv   = (float*)(ws + off); off += (size_t)BATCH * N_IN * 4;
    float* prefA = (float*)(ws + off); off += (size_t)MROWS * OUTF * 4;
    float* prefB = (float*)(ws + off); off += (size_t)MROWS * OUTF * 4;
    float* ZA    = (float*)(ws + off); off += (size_t)MROWS * KP * 4;
    float* ZB    = (float*)(ws + off); off += (size_t)MROWS * KP * 4;
    float* Wp    = (float*)(ws + off); off += (size_t)OUTF * KP * 4;

    // 1) sort + aux arrays: one WG (8 wave32) per batch row
    sort_prep_kernel<<<BATCH, 256, 0, stream>>>(x, pulse, t_s, ord, rank, zv, ztv);

    // 2) build branch-free GEMM operands
    {
        const int nz = MROWS * KP;
        build_z_kernel<<<(nz + 255) / 256, 256, 0, stream>>>(rank, zv, ztv, ZA, ZB);
        const int nw = OUTF * KP;
        build_w_kernel<<<(nw + 255) / 256, 256, 0, stream>>>(W, Wp);
    }

    // 3) carry GEMM: 64x32 = 2048 wave tiles, 4 waves per block
    {
        const int tiles = (MROWS / 16) * (OUTF / 16);
        prefix_gemm_kernel<<<tiles / 4, 128, 0, stream>>>(Wp, ZA, ZB, prefA, prefB);
    }

    // 4) init first-fire slots
    init_kernel<<<(BATCH * OUTF + 255) / 256, 256, 0, stream>>>(packed);

    // 5) parallel chunk refinement
    {
        const int total = BATCH * OUTF * NCHUNK;
        refine_kernel<<<(total + 255) / 256, 256, 0, stream>>>(W, t_s, ord, prefA, prefB, packed);
    }

    // 6) decode
    finalize_kernel<<<(BATCH * OUTF + 255) / 256, 256, 0, stream>>>(packed, out);
}